// BFM_75325136437570
// MI455X (gfx1250) — compile-verified
//
#include <hip/hip_runtime.h>
#include <math.h>

// Problem constants (from the reference)
#define N_U   1000000      // users
#define M_T   500000       // targets / basket items
#define K_DIM 64           // factor dim
#define P_TOT (N_U + 2 * M_T)

// Workspace layout (floats)
//  [0..63]    u_vec accumulator
//  [64..127]  t_vec accumulator
//  [128..191] sum_b accumulator
//  [192]      sum_b_sq accumulator
//  [193]      bias accumulator
#define WS_SQ   192
#define WS_BIAS 193

typedef __attribute__((ext_vector_type(2))) float v2f;
typedef __attribute__((ext_vector_type(8))) float v8f;

// ---------------------------------------------------------------------------
// Kernel 1: zero the accumulator workspace (graph-replay safe re-init)
// ---------------------------------------------------------------------------
__global__ void fm_zero_ws(float* __restrict__ ws) {
    ws[threadIdx.x] = 0.0f;   // 256 threads cover the whole accumulator region
}

// ---------------------------------------------------------------------------
// Kernel 2: sparse scan of x.
// Reads x with coalesced float4 loads (8 MB total ~ 0.35 us @ 23.3 TB/s).
// Threads that hit a nonzero gather the corresponding K=64 embedding row and
// scatter val * row into the workspace accumulators with f32 global atomics.
// Only ~66 nonzeros exist -> ~4.3K atomics total, negligible.
// ---------------------------------------------------------------------------
__global__ void fm_scan(const float4* __restrict__ x4,
                        const float*  __restrict__ w_bias,
                        const float*  __restrict__ u_V,
                        const float*  __restrict__ t_V,
                        const float*  __restrict__ b_V,
                        float*        __restrict__ ws) {
    const int t = blockIdx.x * blockDim.x + threadIdx.x;
    if (t >= P_TOT / 4) return;

    const float4 xv = x4[t];
    const float v[4] = {xv.x, xv.y, xv.z, xv.w};

#pragma unroll
    for (int j = 0; j < 4; ++j) {
        const float val = v[j];
        if (val != 0.0f) {
            const int i = t * 4 + j;
            // bias: only nonzero entries contribute to dot(x, w_bias)
            atomicAdd(&ws[WS_BIAS], val * w_bias[i]);

            const float* V;
            float*       acc;
            int          row;
            bool         is_basket = false;
            if (i < N_U) {
                V = u_V;  acc = ws + 0;   row = i;
            } else if (i < N_U + M_T) {
                V = t_V;  acc = ws + 64;  row = i - N_U;
            } else {
                V = b_V;  acc = ws + 128; row = i - N_U - M_T;
                is_basket = true;
            }
            const float* r = V + (size_t)row * K_DIM;
            float sq = 0.0f;
#pragma unroll 8
            for (int k = 0; k < K_DIM; ++k) {
                const float w = r[k];
                atomicAdd(&acc[k], val * w);
                sq += w * w;
            }
            if (is_basket) atomicAdd(&ws[WS_SQ], val * sq);
        }
    }
}

// ---------------------------------------------------------------------------
// Kernel 3: finalize. One wave (32 lanes, EXEC all-ones as WMMA requires).
// Builds G = [u_vec; t_vec; sum_b] (3x64, zero-padded to a full 16x64 LDS
// tile) and computes the Gram matrix G*G^T with 16 chained
// V_WMMA_F32_16X16X4_F32 steps (f32 WMMA, exact arithmetic). The zero-padded
// tile makes every lane's operand fetch an unconditional, 8-byte-aligned
// ds_load_b64 -- no EXEC-predicated loads, so the WMMA chain can be
// software-pipelined by the scheduler. For Gram products, the documented
// A(16x4) and B(4x16) per-lane register layouts coincide, so the same v2f
// feeds both operands.
//   Gram[0][1]=u.t  Gram[1][2]=t.b  Gram[0][2]=u.b  Gram[2][2]=b.b
// ---------------------------------------------------------------------------
__global__ void fm_finalize(const float* __restrict__ ws,
                            const float* __restrict__ w_0,
                            const float* __restrict__ delta,
                            float*       __restrict__ out) {
    __shared__ float gmat[16 * K_DIM];  // 4 KB: G rows 0..2, rows 3..15 zero
    __shared__ float gram[16 * 16];     // D readback

    const int l = threadIdx.x;          // 0..31, one full wave

    // Zero the whole 16x64 tile, then overwrite rows 0..2 with accumulators.
    for (int j = l; j < 16 * K_DIM; j += 32) gmat[j] = 0.0f;
    __syncthreads();
    for (int j = l; j < 3 * K_DIM; j += 32) gmat[j] = ws[j];
    __syncthreads();

    const int m  = l & 15;              // matrix row handled by this lane
    const int kh = l >> 4;              // 0 -> K pair {0,1}, 1 -> K pair {2,3}

    // Per-lane operand pointer: two consecutive floats, 8-byte aligned.
    const float* base = &gmat[m * K_DIM + kh * 2];

    v8f c = {};                         // zero-init accumulator
#pragma unroll
    for (int kk = 0; kk < 16; ++kk) {   // 16 chunks of K=4 cover K_DIM=64
        const v2f a = *(const v2f*)(base + kk * 4);   // ds_load_b64
        // D = A * B + C with B == A (per-lane) for the Gram product.
        c = __builtin_amdgcn_wmma_f32_16x16x4_f32(
                /*neg_a=*/false, a, /*neg_b=*/false, a,
                /*c_mod=*/(short)0, c, /*reuse_a=*/false, /*reuse_b=*/false);
    }

    // C/D layout: VGPR v, lane l -> M = v + 8*(l>=16), N = l%16
#pragma unroll
    for (int v = 0; v < 8; ++v) {
        const int Mrow = v + 8 * kh;
        gram[Mrow * 16 + m] = c[v];
    }
    __syncthreads();

    if (l == 0) {
        const float u_t = gram[0 * 16 + 1];
        const float u_b = gram[0 * 16 + 2];
        const float t_b = gram[1 * 16 + 2];
        const float b_b = gram[2 * 16 + 2];
        const float bs  = 0.5f * (b_b - ws[WS_SQ]);
        // GAMMA = (1,1,1,1)
        const float y = w_0[0] + ws[WS_BIAS] + u_t + t_b + bs + u_b;
        const float z = y * delta[0];
        out[0] = 1.0f / (1.0f + __expf(-z));
    }
}

// ---------------------------------------------------------------------------
// Launch
// ---------------------------------------------------------------------------
extern "C" void kernel_launch(void* const* d_in, const int* in_sizes, int n_in,
                              void* d_out, int out_size, void* d_ws, size_t ws_size,
                              hipStream_t stream) {
    (void)in_sizes; (void)n_in; (void)out_size; (void)ws_size;

    const float* x      = (const float*)d_in[0];  // (P,)
    const float* delta  = (const float*)d_in[1];  // (1,)
    //            pmi   =               d_in[2];  // unused by reference
    const float* w_0    = (const float*)d_in[3];  // (1,)
    const float* w_bias = (const float*)d_in[4];  // (P,1)
    const float* u_V    = (const float*)d_in[5];  // (N,K)
    const float* t_V    = (const float*)d_in[6];  // (M,K)
    const float* b_V    = (const float*)d_in[7];  // (M,K)

    float* ws  = (float*)d_ws;
    float* out = (float*)d_out;

    fm_zero_ws<<<1, 256, 0, stream>>>(ws);

    const int nthreads = P_TOT / 4;                 // float4 per thread
    const int nblocks  = (nthreads + 255) / 256;
    fm_scan<<<nblocks, 256, 0, stream>>>((const float4*)x, w_bias,
                                         u_V, t_V, b_V, ws);

    fm_finalize<<<1, 32, 0, stream>>>(ws, w_0, delta, out);
}